// Cache_3908420239588
// MI455X (gfx1250) — compile-verified
//
#include <hip/hip_runtime.h>

#define GRID_NP 128
#define GRID_ND 64
#define CH_D    8
#define SCALE_F 3.0f
#define BETA_FLOATS (GRID_ND * GRID_ND * CH_D)   // 32768 floats = 128 KB

__device__ __forceinline__ float fast_rcp(float v) {
#if defined(__AMDGCN__) && __has_builtin(__builtin_amdgcn_rcpf)
    return __builtin_amdgcn_rcpf(v);        // v_rcp_f32
#else
    return 1.0f / v;
#endif
}

__device__ __forceinline__ float fast_sigmoid(float v) {
#if defined(__AMDGCN__) && __has_builtin(__builtin_amdgcn_tanhf)
    // gfx1250 hardware tanh: sigmoid(x) = 0.5*tanh(x/2) + 0.5
    return fmaf(0.5f, __builtin_amdgcn_tanhf(v * 0.5f), 0.5f);
#else
    return fast_rcp(1.0f + __expf(-v));
#endif
}

__device__ __forceinline__ int clampi(int v, int lo, int hi) {
    return v < lo ? lo : (v > hi ? hi : v);
}

__global__ __launch_bounds__(1024) void nerf_fused_kernel(
    const float* __restrict__ x,
    const float* __restrict__ dvec,
    const float* __restrict__ sigma_uvw,
    const float* __restrict__ beta,
    float* __restrict__ out,
    int N)
{
    // 128 KB static LDS: gfx1250 WGP has 320 KB (2 blocks/WGP resident).
    __shared__ float s_beta[BETA_FLOATS];

    const int tid = threadIdx.x;
    const int bs  = blockDim.x;   // 1024

    // ---------------- stage beta table into LDS (async path) ----------------
#if defined(__AMDGCN__) && __has_builtin(__builtin_amdgcn_global_load_async_to_lds_b128)
    {
        typedef int v4i __attribute__((vector_size(16)));
        typedef __attribute__((address_space(1))) v4i gv4i;
        typedef __attribute__((address_space(3))) v4i lv4i;
        typedef __attribute__((address_space(1))) char gchar;
        typedef __attribute__((address_space(3))) char lchar;
        gchar* gp = (gchar*)beta;
        lchar* lp = (lchar*)s_beta;
        #pragma unroll 2
        for (int i = tid; i < BETA_FLOATS / 4; i += bs) {
            __builtin_amdgcn_global_load_async_to_lds_b128(
                (gv4i*)(gp + (size_t)i * 16),
                (lv4i*)(lp + (size_t)i * 16),
                0, 0);
        }
    }
#else
    for (int i = tid; i < BETA_FLOATS / 4; i += bs) {
        ((float4*)s_beta)[i] = ((const float4*)beta)[i];
    }
#endif

    // ------------- per-point work independent of LDS (overlaps copy) --------
    const int  n      = blockIdx.x * bs + tid;
    const bool active = (n < N);

    float x0 = 0.f, x1 = 0.f, x2 = 0.f, d0 = 0.f, d1 = 0.f;
    if (active) {
        const float* xp = x + 3 * (size_t)n;
        x0 = __builtin_nontemporal_load(xp + 0);
        x1 = __builtin_nontemporal_load(xp + 1);
        x2 = __builtin_nontemporal_load(xp + 2);
        const float* dp = dvec + 3 * (size_t)n;
        d0 = __builtin_nontemporal_load(dp + 0);
        d1 = __builtin_nontemporal_load(dp + 1);
        // reference uses only didx[:,0] and didx[:,1]; d[:,2] is never read.
    }

    const float half = SCALE_F * 0.5f;
    const bool mask = active & (fabsf(x0) < half) & (fabsf(x1) < half) & (fabsf(x2) < half);

    float s0 = 0.f;
    float sv[24];
    #pragma unroll
    for (int j = 0; j < 24; ++j) sv[j] = 0.f;
    int brow_idx = 0;

    if (mask) {
        const float inv = (float)GRID_NP / SCALE_F;     // 128/3
        int p0 = clampi((int)(x0 * inv + 0.5f * (float)GRID_NP), 0, GRID_NP - 1);
        int p1 = clampi((int)(x1 * inv + 0.5f * (float)GRID_NP), 0, GRID_NP - 1);
        int p2 = clampi((int)(x2 * inv + 0.5f * (float)GRID_NP), 0, GRID_NP - 1);
        size_t cell = ((size_t)p0 * GRID_NP + (size_t)p1) * GRID_NP + (size_t)p2;
        const float* srow = sigma_uvw + cell * 25;
        s0 = srow[0];
        #pragma unroll
        for (int j = 0; j < 24; ++j) sv[j] = srow[1 + j];

        int q0 = clampi((int)(d0 * (float)GRID_ND), 0, GRID_ND - 1);
        int q1 = clampi((int)(d1 * (float)GRID_ND), 0, GRID_ND - 1);
        brow_idx = (q0 * GRID_ND + q1) * CH_D;
    }

    // ------------- make LDS copy visible, then consume it -------------------
#if defined(__AMDGCN__) && __has_builtin(__builtin_amdgcn_global_load_async_to_lds_b128)
  #if __has_builtin(__builtin_amdgcn_s_wait_asynccnt)
    __builtin_amdgcn_s_wait_asynccnt(0);
  #else
    asm volatile("s_wait_asynccnt 0" ::: "memory");
  #endif
#endif
    __syncthreads();

    float c0 = 0.f, c1 = 0.f, c2 = 0.f, sgm = 0.f;
    if (mask) {
        // beta row from LDS (two ds_load_b128)
        const float4* brow = (const float4*)(s_beta + brow_idx);
        float4 ba = brow[0];
        float4 bb = brow[1];
        float b[8] = { ba.x, ba.y, ba.z, ba.w, bb.x, bb.y, bb.z, bb.w };

        float bm = b[0];
        #pragma unroll
        for (int j = 1; j < 8; ++j) bm = fmaxf(bm, b[j]);

        float e[8];
        float denom = 0.f;
        #pragma unroll
        for (int j = 0; j < 8; ++j) { e[j] = __expf(b[j] - bm); denom += e[j]; }

        float a0 = 0.f, a1 = 0.f, a2 = 0.f;
        #pragma unroll
        for (int j = 0; j < 8; ++j) {
            float ej = e[j];
            a0 = fmaf(ej, fast_sigmoid(sv[j]),      a0);
            a1 = fmaf(ej, fast_sigmoid(sv[8 + j]),  a1);
            a2 = fmaf(ej, fast_sigmoid(sv[16 + j]), a2);
        }
        float rden = fast_rcp(denom);               // v_rcp_f32
        c0 = a0 * rden;
        c1 = a1 * rden;
        c2 = a2 * rden;

        // numerically-stable softplus
        sgm = fmaxf(s0, 0.f) + __logf(1.f + __expf(-fabsf(s0)));
    }

    if (active) {
        float* op = out + 3 * (size_t)n;
        __builtin_nontemporal_store(c0, op + 0);
        __builtin_nontemporal_store(c1, op + 1);
        __builtin_nontemporal_store(c2, op + 2);
        __builtin_nontemporal_store(sgm, out + 3 * (size_t)N + (size_t)n);
    }
}

extern "C" void kernel_launch(void* const* d_in, const int* in_sizes, int n_in,
                              void* d_out, int out_size, void* d_ws, size_t ws_size,
                              hipStream_t stream) {
    (void)n_in; (void)d_ws; (void)ws_size; (void)out_size;
    const float* x    = (const float*)d_in[0];
    const float* d    = (const float*)d_in[1];
    const float* sig  = (const float*)d_in[2];
    const float* beta = (const float*)d_in[3];
    float* out = (float*)d_out;

    const int N = in_sizes[0] / 3;           // 2,097,152
    const int block = 1024;                  // 32 wave32 waves; 1024 pts amortize LDS stage
    const int grid  = (N + block - 1) / block;

    nerf_fused_kernel<<<grid, block, 0, stream>>>(x, d, sig, beta, out, N);
}